// StandardAngular_19164144074833
// MI455X (gfx1250) — compile-verified
//
#include <hip/hip_runtime.h>
#include <math.h>

#define TPB   256
#define PITCH 36   // floats per pair in LDS: 32 payload + 4 pad (bank-conflict-free b128)

#ifdef __has_builtin
# if __has_builtin(__builtin_amdgcn_global_store_async_from_lds_b128)
#  define USE_ASYNC_BUILTIN 1
# endif
# if __has_builtin(__builtin_amdgcn_s_wait_asynccnt)
#  define HAVE_WAIT_ASYNC_BUILTIN 1
# endif
#endif

// Types matching the builtin prototype (confirmed by round-1 diagnostic):
//   param0: v4i addrspace(1)*, param1: v4i addrspace(3)*
typedef int v4i __attribute__((vector_size(16)));
typedef __attribute__((address_space(1))) v4i gv4i;
typedef __attribute__((address_space(3))) v4i lv4i;

__global__ void __launch_bounds__(TPB)
angular_kernel(const float* __restrict__ vec,   // (2, P, 3)
               const float* __restrict__ pEta,  // (1,)
               const float* __restrict__ pZeta, // (1,)
               const float* __restrict__ pShfA, // (8,)
               const float* __restrict__ pShfZ, // (4,)
               float* __restrict__ out,         // (P, 32)
               int P)
{
    __shared__ float lds[TPB * PITCH];           // 36 KB
    const int tid = threadIdx.x;
    const long long p0 = (long long)blockIdx.x * TPB;
    const long long p  = p0 + tid;
    const long long pc = (p < (long long)P) ? p : (long long)(P - 1); // clamp tail reads

    const float eta  = pEta[0];
    const float zeta = pZeta[0];

    const float* a = vec + pc * 3;
    const float* b = vec + (long long)P * 3 + pc * 3;
    float ax = a[0], ay = a[1], az = a[2];
    float bx = b[0], by = b[1], bz = b[2];

    // native v_sqrt_f32 / v_rcp_f32: inputs are well-scaled norms, denominator clamped
    float d1  = __builtin_amdgcn_sqrtf(fmaf(ax, ax, fmaf(ay, ay, az * az)));
    float d2  = __builtin_amdgcn_sqrtf(fmaf(bx, bx, fmaf(by, by, bz * bz)));
    float dot = fmaf(ax, bx, fmaf(ay, by, az * bz));

    // angle = acos(0.95 * dot / max(d1*d2, 1e-10)); keep (cos, sin) instead of angle
    float ca = 0.95f * dot * __builtin_amdgcn_rcpf(fmaxf(d1 * d2, 1e-10f));
    float sa = __builtin_amdgcn_sqrtf(fmaxf(1.0f - ca * ca, 0.0f)); // sin(acos(ca)) >= 0

    const float kf = 3.14159265358979323846f / 3.5f; // pi / CUTOFF
    float fc1 = fmaf(0.5f, __cosf(d1 * kf), 0.5f);
    float fc2 = fmaf(0.5f, __cosf(d2 * kf), 0.5f);
    float pre = 2.0f * fc1 * fc2;                    // 2 * fcj_prod
    float dmean = 0.5f * (d1 + d2);

    // factor1[z] = (0.5*(1+cos(angle - ShfZ[z])))^zeta via angle-addition identity;
    // x^zeta = v_exp_f32(zeta * v_log_f32(x)); x in [0,1], x=0 -> -inf -> 0.
    float f1[4];
#pragma unroll
    for (int z = 0; z < 4; ++z) {
        float sz, cz;
        __sincosf(pShfZ[z], &sz, &cz);
        float x = 0.5f * (1.0f + fmaf(ca, cz, sa * sz));
        f1[z] = __builtin_amdgcn_exp2f(zeta * __builtin_amdgcn_logf(x));
    }

    // factor2[a] * pre, write 32 outputs for this pair into padded LDS row
    float* row = &lds[tid * PITCH];
#pragma unroll
    for (int ai = 0; ai < 8; ++ai) {
        float dd = dmean - pShfA[ai];
        float f2 = pre * __expf(-eta * dd * dd);
        float4 v = make_float4(f2 * f1[0], f2 * f1[1], f2 * f1[2], f2 * f1[3]);
        *(float4*)(row + ai * 4) = v;                // ds_store_b128, conflict-free (pitch 36)
    }

    __syncthreads();

    // Stream the block's 32 KB output tile: fully coalesced 16 B/lane stores.
    const bool fullBlock = (p0 + TPB) <= (long long)P;
    float* gbase = out + p0 * 32;

    if (fullBlock) {
#pragma unroll
        for (int j = 0; j < 8; ++j) {
            int g  = j * TPB + tid;                  // float4 slot in tile, 0..2047
            int pl = g >> 3;                         // pair within block
            int c  = g & 7;                          // float4 chunk within pair
            float* lsrc = &lds[pl * PITCH + c * 4];  // de-pad via address math
#if defined(USE_ASYNC_BUILTIN)
            __builtin_amdgcn_global_store_async_from_lds_b128(
                (gv4i*)(gbase + (long long)g * 4), (lv4i*)lsrc, 0, 0);
#else
            unsigned voff = (unsigned)g * 16u;
            unsigned ldsa = (unsigned)(unsigned long long)lsrc; // low 32 bits = LDS offset
            asm volatile("global_store_async_from_lds_b128 %0, %1, %2"
                         :: "v"(voff), "v"(ldsa), "s"(gbase) : "memory");
#endif
        }
#if defined(HAVE_WAIT_ASYNC_BUILTIN)
        __builtin_amdgcn_s_wait_asynccnt(0);
#else
        asm volatile("s_wait_asynccnt 0x0" ::: "memory");
#endif
    } else {
        // tail block: guarded plain stores (still LDS-coalesced order)
#pragma unroll
        for (int j = 0; j < 8; ++j) {
            int g  = j * TPB + tid;
            int pl = g >> 3;
            int c  = g & 7;
            if (p0 + pl < (long long)P) {
                float4 v = *(const float4*)&lds[pl * PITCH + c * 4];
                *(float4*)(gbase + (long long)g * 4) = v;
            }
        }
    }
}

extern "C" void kernel_launch(void* const* d_in, const int* in_sizes, int n_in,
                              void* d_out, int out_size, void* d_ws, size_t ws_size,
                              hipStream_t stream) {
    const float* vec  = (const float*)d_in[0];  // (2, P, 3) fp32
    const float* eta  = (const float*)d_in[1];  // (1,)
    const float* zeta = (const float*)d_in[2];  // (1,)
    const float* shfA = (const float*)d_in[3];  // (8,)
    const float* shfZ = (const float*)d_in[4];  // (4,)
    float* out = (float*)d_out;                 // (P, 32) fp32

    int P = in_sizes[0] / 6;
    int blocks = (P + TPB - 1) / TPB;
    angular_kernel<<<blocks, TPB, 0, stream>>>(vec, eta, zeta, shfA, shfZ, out, P);
}